// SimpleGaussianSplatting_43138651521248
// MI455X (gfx1250) — compile-verified
//
#include <hip/hip_runtime.h>

// CDNA5 / gfx1250. wave32. V_WMMA_F32_16X16X4_F32 computes 16x16 tiles of
// ||p||^2 - 2 p.c in one matrix op (A row = {-2px,-2py,-2pz,|p|^2},
// B col = {cx,cy,cz,1}); per-ray ||c||^2 is added per lane afterwards.

typedef __attribute__((ext_vector_type(2))) float v2f;
typedef __attribute__((ext_vector_type(8))) float v8f;

#define KSEL 10
#define WAVES_PER_BLOCK 8

__device__ __forceinline__ float sigmoid_f(float x) {
    return 1.0f / (1.0f + __expf(-x));
}

// Sorted-ascending insert of (sq, pi) into best[0..K-1]/bidx[0..K-1].
__device__ __forceinline__ void topk_insert(float sq, int pi,
                                            float best[KSEL], int bidx[KSEL]) {
    if (sq < best[KSEL - 1]) {
        best[KSEL - 1] = sq;
        bidx[KSEL - 1] = pi;
#pragma unroll
        for (int j = KSEL - 1; j > 0; --j) {
            if (best[j] < best[j - 1]) {
                float td = best[j]; best[j] = best[j - 1]; best[j - 1] = td;
                int   ti = bidx[j]; bidx[j] = bidx[j - 1]; bidx[j - 1] = ti;
            }
        }
    }
}

__global__ __launch_bounds__(32 * WAVES_PER_BLOCK)
void gsplat_topk_wmma_kernel(const float* __restrict__ rays_o,
                             const float* __restrict__ rays_d,
                             const float* __restrict__ xyz,
                             const float* __restrict__ fdc,
                             const float* __restrict__ opac,
                             float* __restrict__ out,
                             int N) {
    const int tid  = threadIdx.x;
    const int wave = tid >> 5;         // 0..WAVES_PER_BLOCK-1
    const int lane = tid & 31;         // 0..31
    const int col  = lane & 15;        // D column (ray); also A row (point) to load
    const int hi   = lane >> 4;        // 0: supplies A/B K=0,1, owns D rows 0-7
                                       // 1: supplies A/B K=2,3, owns D rows 8-15
    const int rayBase = blockIdx.x * 16;
    const int r = rayBase + col;

    // Ray center c = o + 3 d for this lane's ray.
    float cx = rays_o[3 * r + 0] + 3.0f * rays_d[3 * r + 0];
    float cy = rays_o[3 * r + 1] + 3.0f * rays_d[3 * r + 1];
    float cz = rays_o[3 * r + 2] + 3.0f * rays_d[3 * r + 2];
    float rn2 = cx * cx + cy * cy + cz * cz;

    // B matrix 4x16 (K x rays): col n = {cx, cy, cz, 1}. Loop-invariant.
    v2f bmat;
    bmat.x = hi ? cz : cx;
    bmat.y = hi ? 1.0f : cy;

    // Per-lane running top-K of squared distances (ascending).
    float best[KSEL];
    int   bidx[KSEL];
#pragma unroll
    for (int k = 0; k < KSEL; ++k) { best[k] = 3.0e38f; bidx[k] = 0; }

    const int rowOff = hi * 8;                  // D rows owned by this lane
    const int nStep  = 16 * WAVES_PER_BLOCK;    // waves stride over the N dim

#pragma unroll 2
    for (int nbase = wave * 16; nbase < N; nbase += nStep) {
        // A matrix 16x4 (points x K): row m = {-2px, -2py, -2pz, |p|^2}
        int pload = nbase + col;
        if (pload > N - 1) pload = N - 1;       // tail guard (N%16==0 here)
        const float* q = xyz + 3 * pload;
        float px = q[0], py = q[1], pz = q[2];
        // Branchless operand build (v_cndmask, no exec churn).
        float a0lo = -2.0f * px, a1lo = -2.0f * py;
        float a0hi = -2.0f * pz, a1hi = px * px + py * py + pz * pz;
        v2f amat;
        amat.x = hi ? a0hi : a0lo;
        amat.y = hi ? a1hi : a1lo;

        v8f acc = {0.f, 0.f, 0.f, 0.f, 0.f, 0.f, 0.f, 0.f};
        // D[m][n] = |p_m|^2 - 2 p_m . c_n
        acc = __builtin_amdgcn_wmma_f32_16x16x4_f32(
            /*neg_a=*/false, amat, /*neg_b=*/false, bmat,
            /*c_mod=*/(short)0, acc, /*reuse_a=*/false, /*reuse_b=*/false);

        // Whole-tile pre-filter: one branch instead of 8 in the common case.
        float m01 = fminf(acc[0], acc[1]);
        float m23 = fminf(acc[2], acc[3]);
        float m45 = fminf(acc[4], acc[5]);
        float m67 = fminf(acc[6], acc[7]);
        float mall = fminf(fminf(m01, m23), fminf(m45, m67));

        if (mall + rn2 < best[KSEL - 1]) {
            if (nbase + 16 <= N) {              // full tile: no index guard
#pragma unroll
                for (int v = 0; v < 8; ++v) {
                    topk_insert(acc[v] + rn2, nbase + rowOff + v, best, bidx);
                }
            } else {                            // ragged tail
#pragma unroll
                for (int v = 0; v < 8; ++v) {
                    int pi = nbase + rowOff + v;
                    if (pi < N) topk_insert(acc[v] + rn2, pi, best, bidx);
                }
            }
        }
    }

    // ---- Merge: each ray has 2*WAVES_PER_BLOCK partial lists ----
    __shared__ float lds_d[32 * WAVES_PER_BLOCK * KSEL];
    __shared__ int   lds_i[32 * WAVES_PER_BLOCK * KSEL];
#pragma unroll
    for (int k = 0; k < KSEL; ++k) {
        lds_d[tid * KSEL + k] = best[k];
        lds_i[tid * KSEL + k] = bidx[k];
    }
    __syncthreads();

    if (tid < 16) {
        float mb[KSEL]; int mi[KSEL];
#pragma unroll
        for (int k = 0; k < KSEL; ++k) { mb[k] = 3.0e38f; mi[k] = 0; }

        for (int w = 0; w < WAVES_PER_BLOCK; ++w) {
#pragma unroll
            for (int h = 0; h < 2; ++h) {
                int src = w * 32 + h * 16 + tid;
#pragma unroll
                for (int k = 0; k < KSEL; ++k) {
                    topk_insert(lds_d[src * KSEL + k], lds_i[src * KSEL + k], mb, mi);
                }
            }
        }

        // Blend: w_k = exp(-0.1 d_k) * sigmoid(op); rgb = sum w*sigmoid(color)
        float wsum = 0.0f, r0 = 0.0f, r1 = 0.0f, r2 = 0.0f;
#pragma unroll
        for (int k = 0; k < KSEL; ++k) {
            int   p = mi[k];
            float d = sqrtf(fmaxf(mb[k], 0.0f));
            float w = __expf(-0.1f * d) * sigmoid_f(opac[p]);
            wsum += w;
            r0 += w * sigmoid_f(fdc[3 * p + 0]);
            r1 += w * sigmoid_f(fdc[3 * p + 1]);
            r2 += w * sigmoid_f(fdc[3 * p + 2]);
        }
        float inv = 1.0f / (wsum + 1e-8f);
        int ro = rayBase + tid;
        out[3 * ro + 0] = r0 * inv;
        out[3 * ro + 1] = r1 * inv;
        out[3 * ro + 2] = r2 * inv;
    }
}

extern "C" void kernel_launch(void* const* d_in, const int* in_sizes, int n_in,
                              void* d_out, int out_size, void* d_ws, size_t ws_size,
                              hipStream_t stream) {
    const float* rays_o = (const float*)d_in[0];
    const float* rays_d = (const float*)d_in[1];
    const float* xyz    = (const float*)d_in[2];
    const float* fdc    = (const float*)d_in[3];  // [N,1,3] flat
    const float* opac   = (const float*)d_in[4];  // [N,1]   flat
    float* out = (float*)d_out;

    const int B = in_sizes[0] / 3;   // 4096
    const int N = in_sizes[2] / 3;   // 100000

    const int nblocks = B / 16;      // one 16-ray tile per block
    hipLaunchKernelGGL(gsplat_topk_wmma_kernel,
                       dim3(nblocks), dim3(32 * WAVES_PER_BLOCK), 0, stream,
                       rays_o, rays_d, xyz, fdc, opac, out, N);
}